// Fusion_loss_23398981829161
// MI455X (gfx1250) — compile-verified
//
#include <hip/hip_runtime.h>

typedef __attribute__((ext_vector_type(2))) float v2f;
typedef __attribute__((ext_vector_type(4))) float v4f;
typedef __attribute__((ext_vector_type(8))) float v8f;
typedef __attribute__((ext_vector_type(4))) int   v4i;

#define DIMK   1024
#define NCLS   64
#define MROWS  (64 * 128)
#define KCHUNK 64
#define NCH    (DIMK / KCHUNK)          // 16 chunks
#define KP     (KCHUNK + 4)             // 68 % 64 == 4 -> conflict-free b128 LDS reads
#define BUFSZ  (NCLS * KP)
#define NSTG   (NCLS * (KCHUNK / 4) / 256)  // async b128 copies per thread per chunk = 4

#if __has_builtin(__builtin_amdgcn_global_load_async_to_lds_b128) && __has_builtin(__builtin_amdgcn_s_wait_asynccnt)
#define HAVE_ASYNC_LDS 1
typedef __attribute__((address_space(1))) v4i as1_v4i;
typedef __attribute__((address_space(3))) v4i as3_v4i;
#endif

// ---------------------------------------------------------------------------
// Kernel 0: precompute ||proto||^2 for each modality into ws[0..191]
// ---------------------------------------------------------------------------
__global__ __launch_bounds__(256) void prep_kernel(const float* __restrict__ p0,
                                                   const float* __restrict__ p1,
                                                   const float* __restrict__ p2,
                                                   float* __restrict__ ws) {
  int t = threadIdx.x;
  if (t < 192) {
    int m = t >> 6, c = t & 63;
    const float* p = (m == 0) ? p0 : (m == 1) ? p1 : p2;
    const float* row = p + (size_t)c * DIMK;
    float s = 0.f;
    for (int k = 0; k < DIMK; k += 4) {
      v4f x = *(const v4f*)(row + k);
      s += x.x * x.x + x.y * x.y + x.z * x.z + x.w * x.w;
    }
    ws[t] = s;
  }
}

#if HAVE_ASYNC_LDS
// issue NSTG async b128 copies of one B chunk (proto[c][kc..kc+KCHUNK)) into LDS buf
__device__ __forceinline__ void stage_chunk(const float* __restrict__ Pm, float* Bs,
                                            int buf, int kc, int tid) {
#pragma unroll
  for (int jj = 0; jj < NSTG; jj++) {
    int j  = tid + jj * 256;
    int c  = j >> 4;                 // KCHUNK/4 == 16 float4 per proto row
    int k4 = (j & 15) << 2;
    __builtin_amdgcn_global_load_async_to_lds_b128(
        (as1_v4i*)(Pm + (size_t)c * DIMK + kc + k4),
        (as3_v4i*)&Bs[buf * BUFSZ + c * KP + k4], 0, 0);
  }
}
#endif

// ---------------------------------------------------------------------------
// One modality: GEMM (WMMA f32 16x16x4, K permuted in slabs of 8) + softmax.
// Double-buffered async B staging: chunk i+1 copies fly behind chunk i compute.
// ---------------------------------------------------------------------------
__device__ __forceinline__ void modality_pass(
    const float* __restrict__ Q, const float* __restrict__ Pm,
    const float* __restrict__ Vm, const float* __restrict__ ws,
    float* Bs, int m, float wgt,
    int tid, int h, int ln, int ty, int yl, int rowbase,
    v8f& pr0, v8f& pr1, v8f& pr2, v8f& pr3, float* rowloss) {

  const float* Aq = Q + (size_t)(rowbase + ln) * DIMK + 4 * h;
  v8f a0 = {}, a1 = {}, a2 = {}, a3 = {};
  float x2p = 0.f;

  for (int ic = 0; ic < NCH; ic++) {
    const int kc = ic * KCHUNK;
    __syncthreads();                       // prev readers of the buffer we stage next
#if HAVE_ASYNC_LDS
    if (ic == 0) stage_chunk(Pm, Bs, 0, 0, tid);
    if (ic + 1 < NCH) {
      stage_chunk(Pm, Bs, (ic + 1) & 1, kc + KCHUNK, tid);
      __builtin_amdgcn_s_wait_asynccnt(NSTG);   // retire current chunk, keep next in flight
    } else {
      __builtin_amdgcn_s_wait_asynccnt(0);
    }
    const float* Bcur = Bs + (ic & 1) * BUFSZ;
#else
    for (int j = tid; j < NCLS * (KCHUNK / 4); j += 256) {
      int c = j >> 4, k4 = (j & 15) << 2;
      *(v4f*)&Bs[c * KP + k4] = *(const v4f*)(Pm + (size_t)c * DIMK + kc + k4);
    }
    const float* Bcur = Bs;
#endif
    __syncthreads();                       // current chunk visible to all waves

    __builtin_prefetch(Aq + kc + KCHUNK, 0, 0);   // stream-ahead on A

#pragma unroll 2
    for (int u = 0; u < KCHUNK; u += 8) {
      // A: one b128 per lane covers ks {u+4h .. u+4h+3} for row ln -> 2 WMMA steps
      v4f a = *(const v4f*)(Aq + kc + u);
      x2p += a.x * a.x + a.y * a.y + a.z * a.z + a.w * a.w;

      const float* brow = &Bcur[ln * KP + u + 4 * h];
      v4f b0 = *(const v4f*)(brow);
      v4f b1 = *(const v4f*)(brow + 16 * KP);
      v4f b2 = *(const v4f*)(brow + 32 * KP);
      v4f b3 = *(const v4f*)(brow + 48 * KP);

      v2f alo = {a.x, a.y}, ahi = {a.z, a.w};
      v2f c0, c1, c2, c3;
      c0 = (v2f){b0.x, b0.y}; c1 = (v2f){b1.x, b1.y};
      c2 = (v2f){b2.x, b2.y}; c3 = (v2f){b3.x, b3.y};
      a0 = __builtin_amdgcn_wmma_f32_16x16x4_f32(false, alo, false, c0, (short)0, a0, false, false);
      a1 = __builtin_amdgcn_wmma_f32_16x16x4_f32(false, alo, false, c1, (short)0, a1, false, false);
      a2 = __builtin_amdgcn_wmma_f32_16x16x4_f32(false, alo, false, c2, (short)0, a2, false, false);
      a3 = __builtin_amdgcn_wmma_f32_16x16x4_f32(false, alo, false, c3, (short)0, a3, false, false);
      c0 = (v2f){b0.z, b0.w}; c1 = (v2f){b1.z, b1.w};
      c2 = (v2f){b2.z, b2.w}; c3 = (v2f){b3.z, b3.w};
      a0 = __builtin_amdgcn_wmma_f32_16x16x4_f32(false, ahi, false, c0, (short)0, a0, false, false);
      a1 = __builtin_amdgcn_wmma_f32_16x16x4_f32(false, ahi, false, c1, (short)0, a1, false, false);
      a2 = __builtin_amdgcn_wmma_f32_16x16x4_f32(false, ahi, false, c2, (short)0, a2, false, false);
      a3 = __builtin_amdgcn_wmma_f32_16x16x4_f32(false, ahi, false, c3, (short)0, a3, false, false);
    }
  }

  // ||x||^2 per row: lane and lane^16 each hold half the K terms for row ln
  float x2full = x2p + __shfl_xor(x2p, 16, 32);
  float x2r[8];
#pragma unroll
  for (int v = 0; v < 8; v++) x2r[v] = __shfl(x2full, v + 8 * h, 32);

  float pc0 = ws[m * 64 + ln],      pc1 = ws[m * 64 + 16 + ln];
  float pc2 = ws[m * 64 + 32 + ln], pc3 = ws[m * 64 + 48 + ln];
  float iv0 = 1.f / Vm[ln],         iv1 = 1.f / Vm[16 + ln];
  float iv2 = 1.f / Vm[32 + ln],    iv3 = 1.f / Vm[48 + ln];

  // logits = (2*xp - x2 - p2) / var
#pragma unroll
  for (int v = 0; v < 8; v++) {
    a0[v] = (2.f * a0[v] - x2r[v] - pc0) * iv0;
    a1[v] = (2.f * a1[v] - x2r[v] - pc1) * iv1;
    a2[v] = (2.f * a2[v] - x2r[v] - pc2) * iv2;
    a3[v] = (2.f * a3[v] - x2r[v] - pc3) * iv3;
  }

#pragma unroll
  for (int v = 0; v < 8; v++) {
    float rmax = fmaxf(fmaxf(a0[v], a1[v]), fmaxf(a2[v], a3[v]));
    rmax = fmaxf(rmax, __shfl_xor(rmax, 1, 32));
    rmax = fmaxf(rmax, __shfl_xor(rmax, 2, 32));
    rmax = fmaxf(rmax, __shfl_xor(rmax, 4, 32));
    rmax = fmaxf(rmax, __shfl_xor(rmax, 8, 32));

    float es = __expf(a0[v] - rmax) + __expf(a1[v] - rmax) +
               __expf(a2[v] - rmax) + __expf(a3[v] - rmax);
    es += __shfl_xor(es, 1, 32);
    es += __shfl_xor(es, 2, 32);
    es += __shfl_xor(es, 4, 32);
    es += __shfl_xor(es, 8, 32);
    float lse = rmax + __logf(es);

    float cand = 0.f;
    if (ln == yl)
      cand = (ty == 0) ? a0[v] : (ty == 1) ? a1[v] : (ty == 2) ? a2[v] : a3[v];
    cand += __shfl_xor(cand, 1, 32);
    cand += __shfl_xor(cand, 2, 32);
    cand += __shfl_xor(cand, 4, 32);
    cand += __shfl_xor(cand, 8, 32);

    rowloss[v] += -wgt * (cand - lse);

    pr0[v] += wgt * __expf(a0[v] - lse);
    pr1[v] += wgt * __expf(a1[v] - lse);
    pr2[v] += wgt * __expf(a2[v] - lse);
    pr3[v] += wgt * __expf(a3[v] - lse);
  }
}

// ---------------------------------------------------------------------------
// Kernel 1: fused 3-modality GEMM + softmax + loss/acc partials
// grid = 64 blocks x 256 threads (8 waves); wave w owns rows blk*128+w*16..+15
// ---------------------------------------------------------------------------
__global__ __launch_bounds__(256) void fusion_kernel(
    const float* __restrict__ q0, const float* __restrict__ p0, const float* __restrict__ v0,
    const float* __restrict__ q1, const float* __restrict__ p1, const float* __restrict__ v1,
    const float* __restrict__ q2, const float* __restrict__ p2, const float* __restrict__ v2,
    float* __restrict__ ws) {
  __shared__ float Bs[2 * BUFSZ];
  __shared__ float redl[8];
  __shared__ float redc[8];

  const int tid  = threadIdx.x;
  const int lane = tid & 31;
  const int h    = lane >> 4;
  const int ln   = lane & 15;
  const int wave = tid >> 5;
  const int rowbase = blockIdx.x * 128 + wave * 16;
  const int y = blockIdx.x;                 // label for all rows of this block
  const int ty = y >> 4, yl = y & 15;

  v8f pr0 = {}, pr1 = {}, pr2 = {}, pr3 = {};
  float rowloss[8];
#pragma unroll
  for (int v = 0; v < 8; v++) rowloss[v] = 0.f;

  modality_pass(q0, p0, v0, ws, Bs, 0, 1.0f, tid, h, ln, ty, yl, rowbase,
                pr0, pr1, pr2, pr3, rowloss);
  modality_pass(q1, p1, v1, ws, Bs, 1, 0.8f, tid, h, ln, ty, yl, rowbase,
                pr0, pr1, pr2, pr3, rowloss);
  modality_pass(q2, p2, v2, ws, Bs, 2, 0.6f, tid, h, ln, ty, yl, rowbase,
                pr0, pr1, pr2, pr3, rowloss);

  // argmax over fused preds + per-wave partial sums
  float lossp = 0.f, correct = 0.f;
#pragma unroll
  for (int v = 0; v < 8; v++) {
    float bv = pr0[v]; int bc = ln;
    if (pr1[v] > bv) { bv = pr1[v]; bc = 16 + ln; }
    if (pr2[v] > bv) { bv = pr2[v]; bc = 32 + ln; }
    if (pr3[v] > bv) { bv = pr3[v]; bc = 48 + ln; }
#pragma unroll
    for (int d = 1; d <= 8; d <<= 1) {
      float ov = __shfl_xor(bv, d, 32);
      int   oc = __shfl_xor(bc, d, 32);
      if (ov > bv || (ov == bv && oc < bc)) { bv = ov; bc = oc; }
    }
    correct += (bc == y) ? 1.f : 0.f;
    lossp   += rowloss[v];
  }
  if (ln != 0) { lossp = 0.f; correct = 0.f; }
  lossp   += __shfl_xor(lossp, 16, 32);
  correct += __shfl_xor(correct, 16, 32);

  if (lane == 0) { redl[wave] = lossp; redc[wave] = correct; }
  __syncthreads();
  if (tid == 0) {
    float L = 0.f, C = 0.f;
    for (int i = 0; i < 8; i++) { L += redl[i]; C += redc[i]; }
    ws[192 + blockIdx.x] = L;
    ws[256 + blockIdx.x] = C;
  }
}

// ---------------------------------------------------------------------------
// Kernel 2: deterministic final reduction -> (loss, acc)
// ---------------------------------------------------------------------------
__global__ void finalize_kernel(const float* __restrict__ ws, float* __restrict__ out) {
  if (threadIdx.x == 0 && blockIdx.x == 0) {
    float L = 0.f, C = 0.f;
    for (int i = 0; i < 64; i++) { L += ws[192 + i]; C += ws[256 + i]; }
    out[0] = L / (float)MROWS;
    out[1] = C / (float)MROWS;
  }
}

extern "C" void kernel_launch(void* const* d_in, const int* in_sizes, int n_in,
                              void* d_out, int out_size, void* d_ws, size_t ws_size,
                              hipStream_t stream) {
  const float* q0 = (const float*)d_in[0];  // TFq_output
  const float* p0 = (const float*)d_in[1];  // TFs_proto
  const float* v0 = (const float*)d_in[2];  // TF_variance
  const float* q1 = (const float*)d_in[3];  // DEq_output
  const float* p1 = (const float*)d_in[4];  // DEs_proto
  const float* v1 = (const float*)d_in[5];  // DE_variance
  const float* q2 = (const float*)d_in[6];  // FFTq_output
  const float* p2 = (const float*)d_in[7];  // FFTs_proto
  const float* v2 = (const float*)d_in[8];  // FFT_variance
  float* ws  = (float*)d_ws;
  float* out = (float*)d_out;

  prep_kernel<<<1, 256, 0, stream>>>(p0, p1, p2, ws);
  fusion_kernel<<<64, 256, 0, stream>>>(q0, p0, v0, q1, p1, v1, q2, p2, v2, ws);
  finalize_kernel<<<1, 32, 0, stream>>>(ws, out);
}